// TransparencyHead_520
// MI455X (gfx1250) — compile-verified
//
#include <hip/hip_runtime.h>
#include <hip/hip_bf16.h>
#include <math.h>
#include <stdint.h>

// Problem constants (match reference)
#define VOCAB   50257
#define MASK_ID 50256
#define THREADS 256
#define CHUNK   4096              // floats per staged chunk (16 KB)
#define NCHUNK  12                // 12*4096 = 49152 floats via async DMA path
#define MAINLEN (NCHUNK * CHUNK)

struct Top3 { float v0, v1, v2; int i0, i1, i2; };

__device__ __forceinline__ void top3_insert(Top3& t, float x, int idx) {
  if (x > t.v2) {
    if (x > t.v0)      { t.v2 = t.v1; t.i2 = t.i1; t.v1 = t.v0; t.i1 = t.i0; t.v0 = x; t.i0 = idx; }
    else if (x > t.v1) { t.v2 = t.v1; t.i2 = t.i1; t.v1 = x;    t.i1 = idx; }
    else               { t.v2 = x;    t.i2 = idx; }
  }
}

// Online softmax-entropy state: m = running max, s = sum e^{x-m}, u = sum x*e^{x-m}
__device__ __forceinline__ void online_update(float& m, float& s, float& u, float x) {
  if (x > m) { float c = __expf(m - x); s *= c; u *= c; m = x; }
  float e = __expf(x - m);
  s += e;
  u = fmaf(x, e, u);
}

__device__ __forceinline__ void combine(float& m, float& s, float& u,
                                        float m2, float s2, float u2) {
  float mn = fmaxf(m, m2);
  float c1 = __expf(m - mn);
  float c2 = __expf(m2 - mn);
  s = s * c1 + s2 * c2;
  u = u * c1 + u2 * c2;
  m = mn;
}

__device__ __forceinline__ float softplus_f(float x) {
  return (x > 20.f) ? x : __logf(1.f + __expf(x));
}

__global__ __launch_bounds__(THREADS)
void entropy_topk_kernel(const int*   __restrict__ input_ids,
                         const float* __restrict__ logits,
                         const float* __restrict__ p_raw_scale,
                         const float* __restrict__ p_raw_centre_neg,
                         const float* __restrict__ p_raw_steep,
                         float* __restrict__ out_idx,    // [rows*4] (indices as float)
                         float* __restrict__ out_probs)  // [rows*4]
{
  __shared__ __align__(16) float lds[2][CHUNK];   // 32 KB double buffer
  __shared__ float red[8][9];                     // per-wave partials

  const int row = blockIdx.x;
  const int tid = threadIdx.x;

  const float* rowp = logits + (size_t)row * VOCAB;

  // Rows are only 4B aligned (V odd); scalar prologue of a in {0..3} elements
  // so the async b128 path sees 16B-aligned global addresses.
  const int a = (int)(((16u - (uint32_t)((size_t)rowp & 15u)) & 15u) >> 2);
  const uint64_t sbase = (uint64_t)(size_t)(rowp + a);          // 16B-aligned base (SGPR pair)
  const uint32_t lds_base = (uint32_t)(size_t)(&lds[0][0]);     // low 32 bits == LDS byte addr

  float m = -INFINITY, s = 0.f, u = 0.f;
  Top3 t; t.v0 = t.v1 = t.v2 = -INFINITY; t.i0 = t.i1 = t.i2 = 0;

  // Stage one 16KB chunk: each of the 8 waves issues 4 x async b128
  // (32 lanes * 16B = 512B per instruction), covering its own slice.
  // TH_LOAD_NT: 824 MB streamed once >> 192 MB L2 -> non-temporal policy
  // avoids thrashing the cache hierarchy with never-reused lines.
  auto stage = [&](int k, int sel) {
    #pragma unroll
    for (int j = 0; j < 4; ++j) {
      const int vp = tid + THREADS * j;                                // float4 slot in chunk
      const uint32_t voff  = (uint32_t)(k * CHUNK * 4 + vp * 16);      // global byte offset
      const uint32_t laddr = lds_base + (uint32_t)(sel * CHUNK * 4 + vp * 16);
      asm volatile("global_load_async_to_lds_b128 %0, %1, %2 th:TH_LOAD_NT"
                   :: "v"(laddr), "v"(voff), "s"(sbase)
                   : "memory");
    }
  };

  // ---- Async double-buffered main stream (ASYNCcnt-tracked DMA) ----
  stage(0, 0);
  for (int k = 0; k < NCHUNK; ++k) {
    const int sel = k & 1;
    if (k + 1 < NCHUNK) {
      stage(k + 1, sel ^ 1);                                   // 8 outstanding now
      asm volatile("s_wait_asynccnt 4" ::: "memory");          // first 4 (this buffer) done
    } else {
      asm volatile("s_wait_asynccnt 0" ::: "memory");          // drain
    }
    #pragma unroll
    for (int j = 0; j < 4; ++j) {
      const int vp = tid + THREADS * j;
      const float4 x4 = *reinterpret_cast<const float4*>(&lds[sel][vp * 4]);
      const int g = a + k * CHUNK + vp * 4;                    // absolute vocab index
      online_update(m, s, u, x4.x); top3_insert(t, x4.x, g + 0);
      online_update(m, s, u, x4.y); top3_insert(t, x4.y, g + 1);
      online_update(m, s, u, x4.z); top3_insert(t, x4.z, g + 2);
      online_update(m, s, u, x4.w); top3_insert(t, x4.w, g + 3);
    }
  }

  // ---- Scalar prologue (alignment) and tail ----
  if (tid < a) {
    float x = rowp[tid];
    online_update(m, s, u, x); top3_insert(t, x, tid);
  }
  for (int g = a + MAINLEN + tid; g < VOCAB; g += THREADS) {
    float x = rowp[g];
    online_update(m, s, u, x); top3_insert(t, x, g);
  }

  // ---- Wave32 xor-shuffle reduction ----
  #pragma unroll
  for (int off = 16; off > 0; off >>= 1) {
    float m2 = __shfl_xor(m, off, 32);
    float s2 = __shfl_xor(s, off, 32);
    float u2 = __shfl_xor(u, off, 32);
    combine(m, s, u, m2, s2, u2);
    float w0 = __shfl_xor(t.v0, off, 32); int j0 = __shfl_xor(t.i0, off, 32);
    float w1 = __shfl_xor(t.v1, off, 32); int j1 = __shfl_xor(t.i1, off, 32);
    float w2 = __shfl_xor(t.v2, off, 32); int j2 = __shfl_xor(t.i2, off, 32);
    top3_insert(t, w0, j0);
    top3_insert(t, w1, j1);
    top3_insert(t, w2, j2);
  }

  const int wave = tid >> 5;
  if ((tid & 31) == 0) {
    red[wave][0] = m;    red[wave][1] = s;    red[wave][2] = u;
    red[wave][3] = t.v0; red[wave][4] = t.v1; red[wave][5] = t.v2;
    red[wave][6] = __int_as_float(t.i0);
    red[wave][7] = __int_as_float(t.i1);
    red[wave][8] = __int_as_float(t.i2);
  }
  __syncthreads();

  if (tid == 0) {
    for (int w = 1; w < 8; ++w) {
      combine(m, s, u, red[w][0], red[w][1], red[w][2]);
      top3_insert(t, red[w][3], __float_as_int(red[w][6]));
      top3_insert(t, red[w][4], __float_as_int(red[w][7]));
      top3_insert(t, red[w][5], __float_as_int(red[w][8]));
    }

    // neg_entropy = sum p*x - log(sum e^x) = u/s - (m + ln s)
    const float neg_entropy = u / s - (m + __logf(s));

    const int  token   = input_ids[row];
    const bool is_mask = (token == MASK_ID);

    const float raw_scale = *p_raw_scale;
    const float raw_cn    = *p_raw_centre_neg;
    const float raw_st    = *p_raw_steep;
    const float scale  = 1.f / (1.f + __expf(-raw_scale));
    const float centre = -softplus_f(raw_cn) - 1e-6f;
    const float steep  =  softplus_f(raw_st) + 1e-6f;
    float lam = scale / (1.f + __expf(-steep * (neg_entropy - centre)));
    if (!is_mask) lam = 0.f;

    // softmax over top-3 logits
    const float e1 = __expf(t.v1 - t.v0);
    const float e2 = __expf(t.v2 - t.v0);
    const float inv = 1.f / (1.f + e1 + e2);
    const float q0 = inv, q1 = e1 * inv, q2 = e2 * inv;

    const int i0 = is_mask ? t.i0 : 0;
    const int i1 = is_mask ? t.i1 : 0;
    const int i2 = is_mask ? t.i2 : 0;
    const float qm = is_mask ? 1.f : 0.f;   // topk_probs zeroed on non-mask rows

    const size_t o = (size_t)row * 4;
    out_idx[o + 0] = (float)token;
    out_idx[o + 1] = (float)i0;
    out_idx[o + 2] = (float)i1;
    out_idx[o + 3] = (float)i2;
    out_probs[o + 0] = 1.f - lam;
    out_probs[o + 1] = lam * q0 * qm;
    out_probs[o + 2] = lam * q1 * qm;
    out_probs[o + 3] = lam * q2 * qm;
  }
}

extern "C" void kernel_launch(void* const* d_in, const int* in_sizes, int n_in,
                              void* d_out, int out_size, void* d_ws, size_t ws_size,
                              hipStream_t stream) {
  (void)n_in; (void)out_size; (void)d_ws; (void)ws_size;
  const int*   input_ids = (const int*)  d_in[0];
  const float* logits    = (const float*)d_in[1];
  const float* rscale    = (const float*)d_in[2];
  const float* rcentre   = (const float*)d_in[3];
  const float* rsteep    = (const float*)d_in[4];

  const int rows = in_sizes[0];            // B*T = 4096
  float* out       = (float*)d_out;
  float* out_idx   = out;                  // [rows*4] indices (as float)
  float* out_probs = out + (size_t)rows * 4;

  entropy_topk_kernel<<<dim3(rows), dim3(THREADS), 0, stream>>>(
      input_ids, logits, rscale, rcentre, rsteep, out_idx, out_probs);
}